// DeepFit_89661737271403
// MI455X (gfx1250) — compile-verified
//
#include <hip/hip_runtime.h>
#include <hip/hip_bf16.h>
#include <math.h>

#define NPTS 1024
#define KNN  40
#define HDIM 64
#define ROWS 16
#define BATCH 128

typedef __attribute__((ext_vector_type(2))) float v2f;
typedef __attribute__((ext_vector_type(4))) float v4f;
typedef __attribute__((ext_vector_type(8))) float v8f;

// -------------------------------------------------------------------------
// Kernel 2: per (batch, 16-row block): WMMA distance stripe in LDS,
// top-40 selection per row, fused 3->64 MLP + maxpool + sigmoid weight.
// -------------------------------------------------------------------------
__global__ __launch_bounds__(128) void deepfit_knn_mlp(
    const float* __restrict__ points,   // [B,3,N]
    const float* __restrict__ W1,       // [3,H]
    const float* __restrict__ b1,       // [H]
    const float* __restrict__ W2,       // [H,1]
    const float* __restrict__ b2,       // [1]
    float* __restrict__ wout)           // [B,N]
{
    // interleaved point storage: bint[i] = {x, y, z, 0} -> branchless b64
    // operand fetch for WMMA A/B and b128 broadcast in phase 2
    __shared__ float bint[NPTS][4];
    __shared__ float xxs[NPTS];
    __shared__ float w1s[3 * HDIM], b1s[HDIM], w2s[HDIM];
    __shared__ float dist[ROWS][NPTS + 4];   // row stride 1028 dwords (16B-mult)

    const int b    = blockIdx.x;
    const int rb   = blockIdx.y * ROWS;
    const int tid  = threadIdx.x;            // 0..127
    const int lane = tid & 31;
    const int wv   = tid >> 5;               // 0..3

    const float* pbase = points + (size_t)b * 3 * NPTS;

    // ---- phase 0: stage points, squared norms, MLP params into LDS ----
    for (int i = tid; i < NPTS; i += 128) {
        float x = pbase[i];
        float y = pbase[NPTS + i];
        float z = pbase[2 * NPTS + i];
        v4f p4 = { x, y, z, 0.0f };
        *(v4f*)&bint[i][0] = p4;
        xxs[i] = x * x + y * y + z * z;
    }
    if (tid < 3 * HDIM) w1s[tid] = W1[tid];
    if (tid < HDIM) { b1s[tid] = b1[tid]; w2s[tid] = W2[tid]; }
    __syncthreads();

    // ---- phase 1: distance stripe via V_WMMA_F32_16X16X4_F32 ----
    const int l15  = lane & 15;
    const int hi   = (lane >> 4) & 1;        // 0: K=0,1 (x,y); 1: K=2,3 (z,0)

    // A operand: 16x4 f32 (K=3 zero-padded), one b64 LDS fetch, branchless
    v2f a = *(const v2f*)&bint[rb + l15][hi * 2];

    for (int ct = wv; ct < NPTS / 16; ct += 4) {
        int col = ct * 16 + l15;
        v2f bb = *(const v2f*)&bint[col][hi * 2];

        v8f c = {};
        c = __builtin_amdgcn_wmma_f32_16x16x4_f32(
                /*neg_a=*/false, a, /*neg_b=*/false, bb,
                /*c_mod=*/(short)0, c, /*reuse_a=*/false, /*reuse_b=*/false);

        float xxc = xxs[col];
#pragma unroll
        for (int r = 0; r < 8; ++r) {
            int m = r + hi * 8;
            // squared distance = |xi|^2 + |xj|^2 - 2 <xi,xj>
            dist[m][col] = xxs[rb + m] + xxc - 2.0f * c[r];
        }
    }
    __syncthreads();

    // ---- phase 2: each wave owns 4 rows: top-40 extraction + fused MLP ----
    for (int rr = 0; rr < 4; ++rr) {
        int m = wv * 4 + rr;
        float pooled0 = 0.0f, pooled1 = 0.0f;   // relu outputs are >= 0

        for (int t = 0; t < KNN; ++t) {
            float best = 3.0e38f;
            int   bidx = 0;
            // lane scans 8 x float4 (32 contiguous-per-vector cols):
            // cols [4*lane + 128*i .. +3], conflict-free, 16B-aligned
#pragma unroll
            for (int i = 0; i < 8; ++i) {
                int c0 = (lane << 2) + (i << 7);
                v4f v = *(const v4f*)&dist[m][c0];
#pragma unroll
                for (int j = 0; j < 4; ++j) {
                    if (v[j] < best) { best = v[j]; bidx = c0 + j; }
                }
            }
            // wave min-reduce (value, index), tie -> smaller index
            for (int off = 16; off > 0; off >>= 1) {
                float ov = __shfl_xor(best, off, 32);
                int   oi = __shfl_xor(bidx, off, 32);
                if (ov < best || (ov == best && oi < bidx)) { best = ov; bidx = oi; }
            }
            // owner lane invalidates the extracted element
            if (((bidx >> 2) & 31) == lane) dist[m][bidx] = 3.0e38f;

            // fused MLP feature for neighbor bidx; lane covers h = lane, lane+32
            v4f pj = *(const v4f*)&bint[bidx][0];
            int h0 = lane, h1 = lane + 32;
            float f0 = pj.x * w1s[h0] + pj.y * w1s[HDIM + h0] + pj.z * w1s[2 * HDIM + h0] + b1s[h0];
            float f1 = pj.x * w1s[h1] + pj.y * w1s[HDIM + h1] + pj.z * w1s[2 * HDIM + h1] + b1s[h1];
            f0 = fmaxf(f0, 0.0f);
            f1 = fmaxf(f1, 0.0f);
            pooled0 = fmaxf(pooled0, f0);
            pooled1 = fmaxf(pooled1, f1);
        }

        // w = sigmoid(pooled . W2 + b2)
        float s = pooled0 * w2s[lane] + pooled1 * w2s[lane + 32];
        for (int off = 16; off > 0; off >>= 1) s += __shfl_xor(s, off, 32);
        if (lane == 0) {
            float ww = 1.0f / (1.0f + __expf(-(s + b2[0])));
            wout[(size_t)b * NPTS + rb + m] = ww;
        }
    }
}

// -------------------------------------------------------------------------
// Kernel 3: per batch: weighted quadratic (order-2 jet) fit -> normal
// -------------------------------------------------------------------------
__global__ __launch_bounds__(256) void deepfit_fit(
    const float* __restrict__ points,   // [B,3,N]
    const float* __restrict__ wbuf,     // [B,N]
    float* __restrict__ out)            // [B,3]
{
    const int b    = blockIdx.x;
    const int tid  = threadIdx.x;
    const int lane = tid & 31;
    const int wv   = tid >> 5;          // 0..7

    const float* pbase = points + (size_t)b * 3 * NPTS;
    const float* wrow  = wbuf + (size_t)b * NPTS;

    __shared__ float red[8 * 27];
    __shared__ float hsh;
    __shared__ int   usew;

    // ---- phase A: h = 0.5*(mean|x| + mean|y|), valid_count ----
    float sax = 0.0f, say = 0.0f, fcnt = 0.0f;
    for (int i = tid; i < NPTS; i += 256) {
        sax  += fabsf(pbase[i]);
        say  += fabsf(pbase[NPTS + i]);
        fcnt += (wrow[i] > 0.001f) ? 1.0f : 0.0f;
    }
    for (int off = 16; off > 0; off >>= 1) {
        sax  += __shfl_xor(sax, off, 32);
        say  += __shfl_xor(say, off, 32);
        fcnt += __shfl_xor(fcnt, off, 32);
    }
    if (lane == 0) { red[wv * 3 + 0] = sax; red[wv * 3 + 1] = say; red[wv * 3 + 2] = fcnt; }
    __syncthreads();
    if (tid == 0) {
        float ax = 0.0f, ay = 0.0f, c = 0.0f;
        for (int i = 0; i < 8; ++i) { ax += red[i * 3]; ay += red[i * 3 + 1]; c += red[i * 3 + 2]; }
        float h = 0.5f * (ax / NPTS + ay / NPTS);
        if (fabsf(h) < 1e-4f) h = 0.1f;
        hsh  = h;
        usew = (c > 18.5f) ? 1 : 0;
    }
    __syncthreads();
    const float h  = hsh;
    const int   uw = usew;

    // ---- phase B: accumulate XtX (upper 21) and XtY (6) ----
    float acc[27];
#pragma unroll
    for (int j = 0; j < 27; ++j) acc[j] = 0.0f;

    for (int i = tid; i < NPTS; i += 256) {
        float x = pbase[i];
        float y = pbase[NPTS + i];
        float z = pbase[2 * NPTS + i];
        float wvv = uw ? wrow[i] : 1.0f;
        float xs = x / h, ys = y / h;
        float p[6] = { xs, ys, xs * xs, ys * ys, xs * ys, 1.0f };
        int t2 = 0;
#pragma unroll
        for (int p1 = 0; p1 < 6; ++p1) {
            float wp = wvv * p[p1];
#pragma unroll
            for (int q = p1; q < 6; ++q) acc[t2++] += wp * p[q];
        }
#pragma unroll
        for (int p1 = 0; p1 < 6; ++p1) acc[21 + p1] += wvv * p[p1] * z;
    }
#pragma unroll
    for (int j = 0; j < 27; ++j)
        for (int off = 16; off > 0; off >>= 1) acc[j] += __shfl_xor(acc[j], off, 32);
    if (lane == 0)
        for (int j = 0; j < 27; ++j) red[wv * 27 + j] = acc[j];
    __syncthreads();

    if (tid == 0) {
        float s[27];
        for (int j = 0; j < 27; ++j) {
            float t = 0.0f;
            for (int i = 0; i < 8; ++i) t += red[i * 27 + j];
            s[j] = t;
        }
        // assemble 6x6 augmented system
        float M[6][7];
        int t2 = 0;
        for (int p1 = 0; p1 < 6; ++p1)
            for (int q = p1; q < 6; ++q) { M[p1][q] = s[t2]; M[q][p1] = s[t2]; ++t2; }
        for (int p1 = 0; p1 < 6; ++p1) M[p1][6] = s[21 + p1];

        // Gauss-Jordan with partial pivoting
        for (int cp = 0; cp < 6; ++cp) {
            int piv = cp; float mv = fabsf(M[cp][cp]);
            for (int r = cp + 1; r < 6; ++r) {
                float av = fabsf(M[r][cp]);
                if (av > mv) { mv = av; piv = r; }
            }
            if (piv != cp)
                for (int c = 0; c < 7; ++c) { float tmp = M[cp][c]; M[cp][c] = M[piv][c]; M[piv][c] = tmp; }
            float d = M[cp][cp];
            if (fabsf(d) < 1e-30f) d = 1e-30f;
            float inv = 1.0f / d;
            for (int c = cp; c < 7; ++c) M[cp][c] *= inv;
            for (int r = 0; r < 6; ++r) {
                if (r == cp) continue;
                float f = M[r][cp];
                for (int c = cp; c < 7; ++c) M[r][c] -= f * M[cp][c];
            }
        }
        float beta[6];
        for (int j = 0; j < 6; ++j) beta[j] = M[j][6];
        float h2 = h * h;
        float dv[6] = { h, h, h2, h2, h2, 1.0f };
        for (int j = 0; j < 6; ++j) beta[j] /= dv[j];

        float nx = -beta[0], ny = -beta[1], nz = 1.0f;
        float rin = rsqrtf(nx * nx + ny * ny + nz * nz);
        out[b * 3 + 0] = nx * rin;
        out[b * 3 + 1] = ny * rin;
        out[b * 3 + 2] = nz * rin;
    }
}

// -------------------------------------------------------------------------
extern "C" void kernel_launch(void* const* d_in, const int* in_sizes, int n_in,
                              void* d_out, int out_size, void* d_ws, size_t ws_size,
                              hipStream_t stream) {
    const float* points = (const float*)d_in[0];
    const float* W1     = (const float*)d_in[1];
    const float* b1     = (const float*)d_in[2];
    const float* W2     = (const float*)d_in[3];
    const float* b2     = (const float*)d_in[4];
    // d_in[5] is k (=40), compile-time constant here

    float* wbuf = (float*)d_ws;        // [B, N] per-point weights (B*N*4 = 512 KB)
    float* out  = (float*)d_out;       // [B, 3]

    dim3 grid2(BATCH, NPTS / ROWS);    // 128 x 64 workgroups
    deepfit_knn_mlp<<<grid2, 128, 0, stream>>>(points, W1, b1, W2, b2, wbuf);

    deepfit_fit<<<BATCH, 256, 0, stream>>>(points, wbuf, out);
}